// CombinedModel_4595615007019
// MI455X (gfx1250) — compile-verified
//
#include <hip/hip_runtime.h>
#include <hip/hip_bf16.h>

// CDNA5 / gfx1250, wave32. Fused CombinedModel (GCN x2 + DNN/BN + head).
//
// Strategy (see analysis): GNN scatter dominates (~1.7GB L2-resident traffic),
// GEMMs are tiny (<2.5 GFLOP total) -> keep fp32 and use V_WMMA_F32_16X16X4_F32
// so the dense math rides the matrix pipes without precision loss.

typedef __attribute__((ext_vector_type(2))) float v2f;
typedef __attribute__((ext_vector_type(8))) float v8f;

#define ATOMIC_FADD(p, v) \
  __hip_atomic_fetch_add((p), (v), __ATOMIC_RELAXED, __HIP_MEMORY_SCOPE_AGENT)

// ---------------------------------------------------------------- utilities
__global__ void k_zero(float* __restrict__ p, int n) {
  int i = blockIdx.x * blockDim.x + threadIdx.x;
  if (i < n) p[i] = 0.0f;
}

// deg[dst] += 1 over edges + self-loops (e >= E -> node e-E)
__global__ void k_deg(const int* __restrict__ ei, int E, int Etot,
                      float* __restrict__ deg) {
  int e = blockIdx.x * blockDim.x + threadIdx.x;
  if (e >= Etot) return;
  int d = (e < E) ? ei[E + e] : (e - E);
  ATOMIC_FADD(&deg[d], 1.0f);
}

// deg -> deg^{-1/2} in place (deg >= 1 due to self-loops)
__global__ void k_rsqrt(float* __restrict__ deg, int n) {
  int i = blockIdx.x * blockDim.x + threadIdx.x;
  if (i < n) deg[i] = rsqrtf(deg[i]);
}

// norm[e] = dinv[src] * dinv[dst]
__global__ void k_norm(const int* __restrict__ ei, const float* __restrict__ dinv,
                       int E, int Etot, float* __restrict__ norm) {
  int e = blockIdx.x * blockDim.x + threadIdx.x;
  if (e >= Etot) return;
  int s = (e < E) ? ei[e] : (e - E);
  int d = (e < E) ? ei[E + e] : (e - E);
  norm[e] = dinv[s] * dinv[d];
}

// ------------------------------------------------- WMMA fp32 GEMM, N==64
// C[M x 64] = A[M x K] @ B[K x 64].  One wave per 16-row tile; 4 accumulators
// cover N=64.  Uses V_WMMA_F32_16X16X4_F32 (fp32 in, fp32 acc).
//
// Fragment layouts per ISA 7.12.2 (f32):
//   A 16x4 : lanes 0-15 hold (K=k,k+1), lanes 16-31 hold (K=k+2,k+3), M = lane&15
//   B 4x16 : VGPR0 -> row K=ka, VGPR1 -> row K=ka+1, N = lane&15 per half
//   C/D    : VGPR v -> M = v (lanes 0-15) / 8+v (lanes 16-31), N = lane&15
__global__ __launch_bounds__(256)
void k_gemm64_wmma(const float* __restrict__ A, const float* __restrict__ B,
                   float* __restrict__ C, int M, int K) {
  const int lane = threadIdx.x & 31;
  const int wave = blockIdx.x * (blockDim.x >> 5) + (threadIdx.x >> 5);
  const int row0 = wave * 16;
  if (row0 >= M) return;                 // wave-uniform: EXEC stays all-1s
  const int half = lane >> 4;            // 0 = lanes 0-15, 1 = lanes 16-31
  const int l16  = lane & 15;
  const int arow = row0 + l16;

  v8f acc0 = {}, acc1 = {}, acc2 = {}, acc3 = {};

  for (int k = 0; k < K; k += 4) {
    const int ka = k + half * 2;
    // A fragment: two consecutive K values for this lane's row (8B aligned)
    v2f a = *(const v2f*)(A + (size_t)arow * K + ka);
    // B fragments for the 4 N-tiles
    const float* Br0 = B + (size_t)ka * 64;        // row K=ka
    const float* Br1 = Br0 + 64;                   // row K=ka+1
    v2f b0, b1, b2, b3;
    b0.x = Br0[l16 +  0]; b0.y = Br1[l16 +  0];
    b1.x = Br0[l16 + 16]; b1.y = Br1[l16 + 16];
    b2.x = Br0[l16 + 32]; b2.y = Br1[l16 + 32];
    b3.x = Br0[l16 + 48]; b3.y = Br1[l16 + 48];

    acc0 = __builtin_amdgcn_wmma_f32_16x16x4_f32(false, a, false, b0, (short)0, acc0, false, false);
    acc1 = __builtin_amdgcn_wmma_f32_16x16x4_f32(false, a, false, b1, (short)0, acc1, false, false);
    acc2 = __builtin_amdgcn_wmma_f32_16x16x4_f32(false, a, false, b2, (short)0, acc2, false, false);
    acc3 = __builtin_amdgcn_wmma_f32_16x16x4_f32(false, a, false, b3, (short)0, acc3, false, false);
  }

  const int crow0 = row0 + half * 8;
#pragma unroll
  for (int v = 0; v < 8; ++v) {
    float* Cr = C + (size_t)(crow0 + v) * 64 + l16;
    Cr[0]  = acc0[v];
    Cr[16] = acc1[v];
    Cr[32] = acc2[v];
    Cr[48] = acc3[v];
  }
}

// ---------------------------------------------- message passing (scatter-add)
// one thread per (edge, feature): coalesced gather of h[src] row, coalesced
// global_atomic_add_f32 into out[dst] row. Feature buffers are L2-resident.
__global__ void k_scatter(const float* __restrict__ h, const float* __restrict__ norm,
                          const int* __restrict__ ei, int E, int Etot,
                          float* __restrict__ out) {
  int idx = blockIdx.x * blockDim.x + threadIdx.x;
  int tot = Etot << 6;                  // Etot * 64 (fits in int: ~1.09e8)
  if (idx >= tot) return;
  int e = idx >> 6, f = idx & 63;
  int s = (e < E) ? ei[e] : (e - E);
  int d = (e < E) ? ei[E + e] : (e - E);
  float v = h[(size_t)s * 64 + f] * norm[e];
  ATOMIC_FADD(&out[(size_t)d * 64 + f], v);
}

// out[i] = relu(in[i] + bias[i & 63])
__global__ void k_bias_relu(const float* __restrict__ in, const float* __restrict__ b,
                            float* __restrict__ out, int n) {
  int i = blockIdx.x * blockDim.x + threadIdx.x;
  if (i < n) out[i] = fmaxf(in[i] + b[i & 63], 0.0f);
}

// gnn_emb[f] = mean_n(in[n][f]) + bc2[f]   (bias folded into reduction)
__global__ void k_gnn_mean(const float* __restrict__ in, const float* __restrict__ bc2,
                           float* __restrict__ emb, int N) {
  __shared__ float red[256];
  int f = blockIdx.x;
  float s = 0.0f;
  for (int r = threadIdx.x; r < N; r += 256) s += in[(size_t)r * 64 + f];
  red[threadIdx.x] = s;
  __syncthreads();
  for (int step = 128; step > 0; step >>= 1) {
    if ((int)threadIdx.x < step) red[threadIdx.x] += red[threadIdx.x + step];
    __syncthreads();
  }
  if (threadIdx.x == 0) emb[f] = red[0] / (float)N + bc2[f];
}

// training-mode BatchNorm1d over batch of 256 + ReLU; 64 threads (1 per column)
__global__ void k_bn_relu(const float* __restrict__ d, const float* __restrict__ b1,
                          const float* __restrict__ gamma, const float* __restrict__ beta,
                          float* __restrict__ out) {
  int j = threadIdx.x;                  // 0..63
  float s = 0.0f, ss = 0.0f;
  for (int r = 0; r < 256; ++r) {
    float v = d[r * 64 + j] + b1[j];
    s += v; ss += v * v;
  }
  float mu  = s * (1.0f / 256.0f);
  float var = ss * (1.0f / 256.0f) - mu * mu;
  float inv = rsqrtf(var + 1e-5f) * gamma[j];
  for (int r = 0; r < 256; ++r) {
    float v = (d[r * 64 + j] + b1[j] - mu) * inv + beta[j];
    out[r * 64 + j] = fmaxf(v, 0.0f);
  }
}

// head: out[r] = ([dnn_r | gnn] @ Wf1 + bf1) @ Wf2 + bf2   (256 threads)
__global__ void k_head(const float* __restrict__ dnn, const float* __restrict__ gemb,
                       const float* __restrict__ Wf1, const float* __restrict__ bf1,
                       const float* __restrict__ Wf2, const float* __restrict__ bf2,
                       float* __restrict__ out) {
  __shared__ float ge[64];
  int r = threadIdx.x;                  // 0..255
  if (r < 64) ge[r] = gemb[r];
  __syncthreads();
  float acc = 0.0f;
  for (int j = 0; j < 64; ++j) {
    float t = bf1[j];
    for (int k = 0; k < 64; ++k) t += dnn[r * 64 + k] * Wf1[k * 64 + j];
    for (int k = 0; k < 64; ++k) t += ge[k] * Wf1[(64 + k) * 64 + j];
    acc += t * Wf2[j];
  }
  out[r] = acc + bf2[0];
}

// ------------------------------------------------------------------ launcher
extern "C" void kernel_launch(void* const* d_in, const int* in_sizes, int n_in,
                              void* d_out, int out_size, void* d_ws, size_t ws_size,
                              hipStream_t stream) {
  const float* x1    = (const float*)d_in[0];   // [256 x 768]
  const float* x2    = (const float*)d_in[1];   // [N x 128]
  const int*   ei    = (const int*)d_in[2];     // [2 x E]
  const float* W1    = (const float*)d_in[3];
  const float* b1    = (const float*)d_in[4];
  const float* gamma = (const float*)d_in[5];
  const float* beta  = (const float*)d_in[6];
  const float* Wc1   = (const float*)d_in[7];
  const float* bc1   = (const float*)d_in[8];
  const float* Wc2   = (const float*)d_in[9];
  const float* bc2   = (const float*)d_in[10];
  const float* Wf1   = (const float*)d_in[11];
  const float* bf1   = (const float*)d_in[12];
  const float* Wf2   = (const float*)d_in[13];
  const float* bf2   = (const float*)d_in[14];
  float* out = (float*)d_out;

  const int N    = in_sizes[1] / 128;   // 100000
  const int E    = in_sizes[2] / 2;     // 1600000
  const int Etot = E + N;
  const int NF   = N * 64;

  // workspace layout (floats)
  float* ws   = (float*)d_ws;
  float* deg  = ws;                     // N (becomes dinv in place)
  float* nrm  = deg + N;                // Etot
  float* bufA = nrm + Etot;             // N*64  (pre-scatter GEMM result)
  float* bufS = bufA + (size_t)NF;      // N*64  (scatter target)
  float* bufG = bufS + (size_t)NF;      // N*64  (post-ReLU layer-1 features)
  float* dbuf = bufG + (size_t)NF;      // 256*64 (DNN linear out)
  float* dnn  = dbuf + 256 * 64;        // 256*64 (post-BN/ReLU)
  float* gemb = dnn + 256 * 64;         // 64

  const int T = 256;
  auto blk = [](int n, int t) { return (n + t - 1) / t; };
  const int gemmBlocks  = blk(blk(N, 16), 8);        // 8 waves/block, 16 rows/wave
  const int gemmBlocksD = blk(blk(256, 16), 8);

  // ---- degrees & symmetric norm (shared by both conv layers)
  k_zero<<<blk(N, T), T, 0, stream>>>(deg, N);
  k_deg<<<blk(Etot, T), T, 0, stream>>>(ei, E, Etot, deg);
  k_rsqrt<<<blk(N, T), T, 0, stream>>>(deg, N);
  k_norm<<<blk(Etot, T), T, 0, stream>>>(ei, deg, E, Etot, nrm);

  // ---- GCN layer 1: h = x2 @ Wc1 ; scatter ; +bc1 ; relu
  k_gemm64_wmma<<<gemmBlocks, T, 0, stream>>>(x2, Wc1, bufA, N, 128);
  k_zero<<<blk(NF, T), T, 0, stream>>>(bufS, NF);
  k_scatter<<<blk(Etot * 64, T), T, 0, stream>>>(bufA, nrm, ei, E, Etot, bufS);
  k_bias_relu<<<blk(NF, T), T, 0, stream>>>(bufS, bc1, bufG, NF);

  // ---- GCN layer 2: h = g @ Wc2 ; scatter ; mean(+bc2) -> gnn_emb
  k_gemm64_wmma<<<gemmBlocks, T, 0, stream>>>(bufG, Wc2, bufA, N, 64);
  k_zero<<<blk(NF, T), T, 0, stream>>>(bufS, NF);
  k_scatter<<<blk(Etot * 64, T), T, 0, stream>>>(bufA, nrm, ei, E, Etot, bufS);
  k_gnn_mean<<<64, 256, 0, stream>>>(bufS, bc2, gemb, N);

  // ---- DNN branch: x1 @ W1 ; BN(batch stats) ; relu
  k_gemm64_wmma<<<gemmBlocksD, T, 0, stream>>>(x1, W1, dbuf, 256, 768);
  k_bn_relu<<<1, 64, 0, stream>>>(dbuf, b1, gamma, beta, dnn);

  // ---- head
  k_head<<<1, 256, 0, stream>>>(dnn, gemb, Wf1, bf1, Wf2, bf2, out);

  (void)n_in; (void)out_size; (void)ws_size;
}